// LightGCNLayer_47425028882704
// MI455X (gfx1250) — compile-verified
//
#include <hip/hip_runtime.h>

// LightGCN layer for MI455X (gfx1250).
// Memory/atomic-bound scatter-gather: WMMA inapplicable (unsorted sparse A,
// AI ~0.25 FLOP/B). Use CDNA5 async global->LDS copies (ASYNCcnt) to pipeline
// the random 512B row gathers, and f32 global atomics for the scatter.

#define DFEAT 128
#define WAVES_PER_BLOCK 8   // 256 threads / wave32

// ---- CDNA5 async global->LDS support (guarded; fallback = plain loads) ----
#if defined(__AMDGCN__) && __has_builtin(__builtin_amdgcn_global_load_async_to_lds_b128)
#define HAVE_ASYNC_LDS 1
#else
#define HAVE_ASYNC_LDS 0
#endif

#if defined(__AMDGCN__) && __has_builtin(__builtin_amdgcn_s_wait_asynccnt)
#define WAIT_ASYNC(n) __builtin_amdgcn_s_wait_asynccnt(n)
#else
#define WAIT_ASYNC(n) asm volatile("s_wait_asynccnt " #n)
#endif

#if HAVE_ASYNC_LDS
typedef int v4i __attribute__((vector_size(16)));
#define GLOBAL_AS __attribute__((address_space(1)))
#define LDS_AS    __attribute__((address_space(3)))

// 16B/lane async copy: global -> LDS, tracked by ASYNCcnt.
__device__ __forceinline__ void async_copy_b128(const float* g, float* l) {
  __builtin_amdgcn_global_load_async_to_lds_b128(
      (GLOBAL_AS v4i*)g, (LDS_AS v4i*)l, 0, 0);
}
#endif

__global__ void zero_f32_kernel(float* __restrict__ p, size_t n) {
  size_t i = (size_t)blockIdx.x * blockDim.x + threadIdx.x;
  size_t stride = (size_t)gridDim.x * blockDim.x;
  for (; i < n; i += stride) p[i] = 0.0f;
}

__global__ void degree_kernel(const int* __restrict__ es, const int* __restrict__ ed,
                              float* __restrict__ degOut, float* __restrict__ degIn,
                              int E) {
  int e = blockIdx.x * blockDim.x + threadIdx.x;
  if (e < E) {
    atomicAdd(degOut + es[e], 1.0f);
    atomicAdd(degIn + ed[e], 1.0f);
  }
}

// One wave per edge. Lane l owns floats [4l, 4l+3] of the 128-float row.
__global__ __launch_bounds__(256) void scatter_kernel(
    const float* __restrict__ h, const int* __restrict__ es,
    const int* __restrict__ ed, const float* __restrict__ degOut,
    float* __restrict__ out, int E) {
  const int lane  = threadIdx.x & 31;
  const int wslot = threadIdx.x >> 5;
  const int wid   = blockIdx.x * WAVES_PER_BLOCK + wslot;
  const int nwaves = gridDim.x * WAVES_PER_BLOCK;

#if HAVE_ASYNC_LDS
  // Double-buffered async gather of the src feature row into LDS.
  __shared__ float buf[WAVES_PER_BLOCK][2][DFEAT];   // 8 KB / block
  int e = wid;
  bool haveCur = (e < E);
  int sCur = 0;
  if (haveCur) {
    sCur = es[e];
    async_copy_b128(h + (size_t)sCur * DFEAT + lane * 4,
                    &buf[wslot][0][lane * 4]);
  }
  int kbuf = 0;
  while (haveCur) {
    const int eNext = e + nwaves;
    const bool haveNext = (eNext < E);
    int sNext = 0;
    if (haveNext) {
      sNext = es[eNext];
      async_copy_b128(h + (size_t)sNext * DFEAT + lane * 4,
                      &buf[wslot][kbuf ^ 1][lane * 4]);
    }
    if (haveNext) { WAIT_ASYNC(1); } else { WAIT_ASYNC(0); }

    const int d = ed[e];
    const float norm = rsqrtf(fmaxf(degOut[sCur], 1.0f));
    const float4 v = *(const float4*)(&buf[wslot][kbuf][lane * 4]);
    float* o = out + (size_t)d * DFEAT + lane * 4;
    atomicAdd(o + 0, v.x * norm);
    atomicAdd(o + 1, v.y * norm);
    atomicAdd(o + 2, v.z * norm);
    atomicAdd(o + 3, v.w * norm);

    e = eNext; sCur = sNext; haveCur = haveNext; kbuf ^= 1;
  }
#else
  // Fallback: direct float4 gather + prefetch of next row.
  for (int e = wid; e < E; e += nwaves) {
    const int s = es[e];
    const int d = ed[e];
    const float norm = rsqrtf(fmaxf(degOut[s], 1.0f));
    const float4 v = *(const float4*)(h + (size_t)s * DFEAT + lane * 4);
    const int eNext = e + nwaves;
    if (eNext < E) {
      const int sn = es[eNext];
      __builtin_prefetch(h + (size_t)sn * DFEAT + lane * 4, 0, 0);
    }
    float* o = out + (size_t)d * DFEAT + lane * 4;
    atomicAdd(o + 0, v.x * norm);
    atomicAdd(o + 1, v.y * norm);
    atomicAdd(o + 2, v.z * norm);
    atomicAdd(o + 3, v.w * norm);
  }
#endif
}

__global__ void scale_kernel(float* __restrict__ out, const float* __restrict__ degIn,
                             size_t n4) {
  size_t i = (size_t)blockIdx.x * blockDim.x + threadIdx.x;
  size_t stride = (size_t)gridDim.x * blockDim.x;
  float4* o4 = (float4*)out;
  for (; i < n4; i += stride) {
    const size_t row = i >> 5;  // (i*4) / 128
    const float norm = rsqrtf(fmaxf(degIn[row], 1.0f));
    float4 v = o4[i];
    v.x *= norm; v.y *= norm; v.z *= norm; v.w *= norm;
    o4[i] = v;
  }
}

extern "C" void kernel_launch(void* const* d_in, const int* in_sizes, int n_in,
                              void* d_out, int out_size, void* d_ws, size_t ws_size,
                              hipStream_t stream) {
  const float* h  = (const float*)d_in[0];
  const int*   es = (const int*)d_in[1];
  const int*   ed = (const int*)d_in[2];
  float*       out = (float*)d_out;

  const int E     = in_sizes[1];
  const int n_src = in_sizes[0] / DFEAT;
  const int n_dst = out_size / DFEAT;

  float* degOut = (float*)d_ws;          // n_src floats
  float* degIn  = degOut + n_src;        // n_dst floats

  zero_f32_kernel<<<256, 256, 0, stream>>>(degOut, (size_t)(n_src + n_dst));
  zero_f32_kernel<<<512, 256, 0, stream>>>(out, (size_t)out_size);
  degree_kernel<<<(E + 255) / 256, 256, 0, stream>>>(es, ed, degOut, degIn, E);
  scatter_kernel<<<2048, 256, 0, stream>>>(h, es, ed, degOut, out, E);
  scale_kernel<<<1024, 256, 0, stream>>>(out, degIn, (size_t)out_size / 4);
}